// PhysicsAwareEmbedding_249108103787
// MI455X (gfx1250) — compile-verified
//
#include <hip/hip_runtime.h>
#include <math.h>

#define NN 100000
#define EE 1600000
#define CC 128

typedef __attribute__((ext_vector_type(2))) float v2f;
typedef __attribute__((ext_vector_type(8))) float v8f;

__device__ __forceinline__ float gelu_exact(float v) {
    // gelu(v, approximate=False) = 0.5*v*(1+erf(v/sqrt(2)))
    return 0.5f * v * (1.0f + erff(v * 0.70710678118654752f));
}

// -----------------------------------------------------------------------------
// lift stage 1: t = gelu(x[:,3:9] @ W1 + b1)   (K=6: VALU, not worth WMMA)
// -----------------------------------------------------------------------------
__global__ __launch_bounds__(256)
void lift1_kernel(const float* __restrict__ x,        // (N,9)
                  const float* __restrict__ w1,       // (6,128)
                  const float* __restrict__ b1,       // (128)
                  float* __restrict__ out)            // (N,128)
{
    int t = blockIdx.x * 256 + threadIdx.x;
    int n = t >> 7;
    int c = t & 127;
    if (n >= NN) return;
    const float* xr = x + (size_t)n * 9 + 3;
    float s = b1[c];
#pragma unroll
    for (int j = 0; j < 6; ++j) s += xr[j] * w1[j * 128 + c];
    out[(size_t)n * 128 + c] = gelu_exact(s);
}

// -----------------------------------------------------------------------------
// Generic f32 WMMA GEMM:  Out = [res +] [gelu](A(N x Cin) @ W(Cin x 128) + b)
// Block = 32 rows x 128 cols; 8 waves; each wave owns TWO 16x16 tiles stacked
// in M so each B fragment is reused by 2 WMMAs (halves B traffic, 1.5 loads
// per v_wmma instead of 2).
// Fragment layouts per CDNA5 ISA 7.12.2:
//   A (16x4):  lanes 0-15 hold M=lane, K = k,k+1 in v0,v1; lanes 16-31 K=k+2,k+3
//   B (4x16):  lanes 0-15 hold N=lane, K = k,k+1;          lanes 16-31 K=k+2,k+3
//   D (16x16): VGPR r -> M = r (lanes 0-15) / r+8 (lanes 16-31), N = lane&15
// -----------------------------------------------------------------------------
__global__ __launch_bounds__(256)
void gemm_wmma_f32(const float* __restrict__ A, int lda,
                   const float* __restrict__ W,          // (Cin,128) row-major
                   const float* __restrict__ bias,       // (128)
                   float* __restrict__ Out, int ldo,
                   const float* __restrict__ Res, int ldr,  // optional residual
                   int Cin, int applyGelu)
{
    const int wave = threadIdx.x >> 5;   // 0..7 -> column tile
    const int lane = threadIdx.x & 31;
    const int hi   = lane >> 4;          // half select
    const int lm   = lane & 15;
    const int row0 = blockIdx.x * 32;    // two stacked 16-row tiles
    const int col0 = wave * 16;
    const int koff = hi * 2;

    const float* Arow0 = A + (size_t)(row0 + lm) * lda + koff;       // tile 0
    const float* Arow1 = Arow0 + (size_t)16 * lda;                   // tile 1
    const float* Wcol  = W + (size_t)koff * 128 + (col0 + lm);       // W[k+koff][n]

    v8f acc0 = {};
    v8f acc1 = {};
    for (int k = 0; k < Cin; k += 4) {
        v2f b, a0, a1;
        b.x  = Wcol[(size_t)k * 128];
        b.y  = Wcol[(size_t)(k + 1) * 128];
        a0.x = Arow0[k];
        a0.y = Arow0[k + 1];
        a1.x = Arow1[k];
        a1.y = Arow1[k + 1];
        acc0 = __builtin_amdgcn_wmma_f32_16x16x4_f32(
            false, a0, false, b, (short)0, acc0, false, false);
        acc1 = __builtin_amdgcn_wmma_f32_16x16x4_f32(
            false, a1, false, b, (short)0, acc1, false, false);
    }

    const float bv = bias[col0 + lm];
    const int orow = row0 + hi * 8;
#pragma unroll
    for (int r = 0; r < 8; ++r) {
        float v0 = acc0[r] + bv;
        float v1 = acc1[r] + bv;
        if (applyGelu) { v0 = gelu_exact(v0); v1 = gelu_exact(v1); }
        size_t oi0 = (size_t)(orow + r) * ldo + col0 + lm;
        size_t oi1 = (size_t)(orow + 16 + r) * ldo + col0 + lm;
        if (Res) {
            v0 += Res[(size_t)(orow + r) * ldr + col0 + lm];
            v1 += Res[(size_t)(orow + 16 + r) * ldr + col0 + lm];
        }
        Out[oi0] = v0;
        Out[oi1] = v1;
    }
}

// -----------------------------------------------------------------------------
// Edge scatter: aggr[row[e], 128+c] += neigh[col[e], c] * ev[e]
// One wave per edge; row/col/ev wave-uniform; 32 lanes x float4 = 128 channels.
// -----------------------------------------------------------------------------
__global__ __launch_bounds__(256)
void scatter_kernel(const float* __restrict__ neigh,   // (N,128)
                    const int* __restrict__ row,
                    const int* __restrict__ col,
                    const float* __restrict__ ev,
                    float* __restrict__ aggr)          // (N,256), right half
{
    int e = blockIdx.x * 8 + (threadIdx.x >> 5);
    if (e >= EE) return;
    int lane = threadIdx.x & 31;
    int r = row[e];
    int c = col[e];
    float v = ev[e];
    const float4 m = *(const float4*)(neigh + (size_t)c * 128 + lane * 4);
    float* dst = aggr + (size_t)r * 256 + 128 + lane * 4;
    atomicAdd(dst + 0, m.x * v);
    atomicAdd(dst + 1, m.y * v);
    atomicAdd(dst + 2, m.z * v);
    atomicAdd(dst + 3, m.w * v);
}

// -----------------------------------------------------------------------------
// LayerNorm over C=128 per row; one wave per row, shuffle reduction (wave32).
// -----------------------------------------------------------------------------
__global__ __launch_bounds__(256)
void lnorm_kernel(const float* __restrict__ h,
                  const float* __restrict__ g,
                  const float* __restrict__ b,
                  float* __restrict__ out)
{
    int n = blockIdx.x * 8 + (threadIdx.x >> 5);
    if (n >= NN) return;
    int lane = threadIdx.x & 31;
    int c = lane * 4;
    const float4 v = *(const float4*)(h + (size_t)n * 128 + c);

    float s = v.x + v.y + v.z + v.w;
#pragma unroll
    for (int o = 16; o; o >>= 1) s += __shfl_xor(s, o, 32);
    float mu = s * (1.0f / 128.0f);

    float4 d;
    d.x = v.x - mu; d.y = v.y - mu; d.z = v.z - mu; d.w = v.w - mu;
    float q = d.x * d.x + d.y * d.y + d.z * d.z + d.w * d.w;
#pragma unroll
    for (int o = 16; o; o >>= 1) q += __shfl_xor(q, o, 32);
    float rs = rsqrtf(q * (1.0f / 128.0f) + 1e-5f);

    float4 o4;
    o4.x = d.x * rs * g[c + 0] + b[c + 0];
    o4.y = d.y * rs * g[c + 1] + b[c + 1];
    o4.z = d.z * rs * g[c + 2] + b[c + 2];
    o4.w = d.w * rs * g[c + 3] + b[c + 3];
    *(float4*)(out + (size_t)n * 128 + c) = o4;
}

// -----------------------------------------------------------------------------
extern "C" void kernel_launch(void* const* d_in, const int* in_sizes, int n_in,
                              void* d_out, int out_size, void* d_ws, size_t ws_size,
                              hipStream_t stream) {
    const float* x       = (const float*)d_in[0];
    const int*   eidx    = (const int*)d_in[1];   // (2,E): row=[0..E), col=[E..2E)
    const float* ev      = (const float*)d_in[2];
    const float* lift_w1 = (const float*)d_in[3];
    const float* lift_b1 = (const float*)d_in[4];
    const float* lift_w2 = (const float*)d_in[5];
    const float* lift_b2 = (const float*)d_in[6];
    const float* self_w  = (const float*)d_in[7];
    const float* self_b  = (const float*)d_in[8];
    const float* neigh_w = (const float*)d_in[9];
    const float* neigh_b = (const float*)d_in[10];
    const float* gate_w1 = (const float*)d_in[11];
    const float* gate_b1 = (const float*)d_in[12];
    const float* gate_w2 = (const float*)d_in[13];
    const float* gate_b2 = (const float*)d_in[14];
    const float* norm_g  = (const float*)d_in[15];
    const float* norm_b  = (const float*)d_in[16];

    float* W0 = (float*)d_ws;                        // h        (N,128)
    float* W1 = W0 + (size_t)NN * 128;               // tmp      (N,128)
    float* W2 = W1 + (size_t)NN * 128;               // concat   (N,256)

    const int gemmGrid = NN / 32;     // 3125 blocks, 32 rows each (divides evenly)
    const int gemmBlk  = 256;         // 8 waves -> 8 col tiles of 16 (Cout=128)

    // lift
    lift1_kernel<<<(NN * 128) / 256, 256, 0, stream>>>(x, lift_w1, lift_b1, W1);
    gemm_wmma_f32<<<gemmGrid, gemmBlk, 0, stream>>>(
        W1, 128, lift_w2, lift_b2, W0, 128, nullptr, 0, 128, 0);

    for (int l = 0; l < 2; ++l) {
        // neighbor = h @ neigh_w + b
        gemm_wmma_f32<<<gemmGrid, gemmBlk, 0, stream>>>(
            W0, 128, neigh_w + (size_t)l * 128 * 128, neigh_b + l * 128,
            W1, 128, nullptr, 0, 128, 0);

        // zero concat buffer, scatter messages into right half
        hipMemsetAsync(W2, 0, (size_t)NN * 256 * sizeof(float), stream);
        scatter_kernel<<<EE / 8, 256, 0, stream>>>(W1, eidx, eidx + EE, ev, W2);

        // self_f = h @ self_w + b  -> left half of concat (ldo = 256)
        gemm_wmma_f32<<<gemmGrid, gemmBlk, 0, stream>>>(
            W0, 128, self_w + (size_t)l * 128 * 128, self_b + l * 128,
            W2, 256, nullptr, 0, 128, 0);

        // u1 = gelu(concat @ gate_w1 + b1)   (Cin = 256)
        gemm_wmma_f32<<<gemmGrid, gemmBlk, 0, stream>>>(
            W2, 256, gate_w1 + (size_t)l * 256 * 128, gate_b1 + l * 128,
            W1, 128, nullptr, 0, 256, 1);

        // h = h + u1 @ gate_w2 + b2
        gemm_wmma_f32<<<gemmGrid, gemmBlk, 0, stream>>>(
            W1, 128, gate_w2 + (size_t)l * 128 * 128, gate_b2 + l * 128,
            W0, 128, W0, 128, 128, 0);
    }

    // layernorm -> output
    lnorm_kernel<<<(NN + 7) / 8, 256, 0, stream>>>(W0, norm_g, norm_b, (float*)d_out);
}